// GNNModelPyG_8564164788849
// MI455X (gfx1250) — compile-verified
//
#include <hip/hip_runtime.h>
#include <math.h>

// ---------------------------------------------------------------------------
// GCN (2-layer) + linear head for MI455X (gfx1250, wave32, WMMA).
//   layer: h = X@W ; agg[dst] += h[src]*dinv[src]*dinv[dst] ; 
//          out = relu(agg + h*dinv^2 + b)
//   head : sigmoid(h2 @ Wc + bc)
// GEMMs use V_WMMA_F32_16X16X4_F32 (fp32 A/B/C, 16x16 tiles, K-step 4).
// ---------------------------------------------------------------------------

typedef __attribute__((ext_vector_type(2))) float v2f;
typedef __attribute__((ext_vector_type(8))) float v8f;

#define HIDC 64

// ---- degree count via f32 atomics ----------------------------------------
__global__ void degree_kernel(const int* __restrict__ dst, float* __restrict__ deg, int E) {
    int e = blockIdx.x * blockDim.x + threadIdx.x;
    if (e < E) atomicAdd(&deg[dst[e]], 1.0f);
}

// ---- deg -> dinv = rsqrt(deg + 1)  (self-loop included) -------------------
__global__ void dinv_kernel(float* __restrict__ d, int N) {
    int n = blockIdx.x * blockDim.x + threadIdx.x;
    if (n < N) d[n] = rsqrtf(d[n] + 1.0f);
}

// ---- GEMM: H[N,OUT] = X[N,K] @ W[K,OUT]  (one wave per 16x16 output tile) -
template <int K, int OUT>
__global__ void gemm_wmma_kernel(const float* __restrict__ X,
                                 const float* __restrict__ W,
                                 float* __restrict__ H,
                                 int n_nodes) {
    const int tid   = blockIdx.x * blockDim.x + threadIdx.x;
    const int wave  = tid >> 5;
    const int lane  = threadIdx.x & 31;
    const int ctiles = OUT >> 4;
    const int ntile = wave / ctiles;
    const int ctile = wave - ntile * ctiles;
    const int nbase = ntile << 4;
    if (nbase >= n_nodes) return;            // wave-uniform: EXEC stays full

    const int m     = lane & 15;             // M (A row) == N (B col) for this lane
    const int khalf = (lane >> 4) << 1;      // lanes 0-15 -> K+0..1, lanes 16-31 -> K+2..3
    const int row   = nbase + m;             // n_nodes % 16 == 0, always valid
    const int cbase = ctile << 4;

    v8f acc = {};
#pragma unroll
    for (int kb = 0; kb < K; kb += 4) {
        v2f a, b;
        a.x = X[row * K + kb + khalf];
        a.y = X[row * K + kb + khalf + 1];
        b.x = W[(kb + khalf) * OUT + cbase + m];
        b.y = W[(kb + khalf + 1) * OUT + cbase + m];
        acc = __builtin_amdgcn_wmma_f32_16x16x4_f32(
            /*neg_a=*/false, a, /*neg_b=*/false, b,
            /*c_mod=*/(short)0, acc, /*reuse_a=*/false, /*reuse_b=*/false);
    }

    // C/D layout: VGPR r -> M = r (lanes 0-15) / r+8 (lanes 16-31), N = lane%16
    const int mrow0 = (lane < 16) ? 0 : 8;
#pragma unroll
    for (int r = 0; r < 8; ++r) {
        H[(nbase + mrow0 + r) * OUT + cbase + m] = acc[r];
    }
}

// ---- edge scatter: warp per edge, lane c handles channels 2c, 2c+1 --------
__global__ void scatter_kernel(const int* __restrict__ src,
                               const int* __restrict__ dst,
                               const float* __restrict__ dinv,
                               const float* __restrict__ h,
                               float* __restrict__ agg, int E) {
    int t    = blockIdx.x * blockDim.x + threadIdx.x;
    int e    = t >> 5;
    int lane = threadIdx.x & 31;
    if (e >= E) return;
    int s = src[e];
    int d = dst[e];
    float norm = dinv[s] * dinv[d];
    const float2* hp = (const float2*)(h + (size_t)s * HIDC);
    float2 hv = hp[lane];                            // global_load_b64
    float* ap = agg + (size_t)d * HIDC + 2 * lane;
    atomicAdd(ap,     hv.x * norm);
    atomicAdd(ap + 1, hv.y * norm);
}

// ---- finalize: out = relu(agg + h*dinv^2 + bias)  (in place over agg) -----
__global__ void finalize_kernel(const float* __restrict__ h,
                                const float* __restrict__ dinv,
                                const float* __restrict__ bias,
                                float* __restrict__ agg_out, int N) {
    int t = blockIdx.x * blockDim.x + threadIdx.x;
    if (t >= N * HIDC) return;
    int n = t >> 6;
    int c = t & (HIDC - 1);
    float di = dinv[n];
    float v  = agg_out[t] + h[t] * (di * di) + bias[c];
    agg_out[t] = fmaxf(v, 0.0f);
}

// ---- head: warp per node, dot64 + sigmoid ---------------------------------
__global__ void head_kernel(const float* __restrict__ h2,
                            const float* __restrict__ Wc,
                            const float* __restrict__ bc,
                            float* __restrict__ out, int N) {
    int t    = blockIdx.x * blockDim.x + threadIdx.x;
    int n    = t >> 5;
    int lane = threadIdx.x & 31;
    if (n >= N) return;
    const float* hp = h2 + (size_t)n * HIDC;
    float acc = hp[lane] * Wc[lane] + hp[lane + 32] * Wc[lane + 32];
#pragma unroll
    for (int off = 16; off > 0; off >>= 1)
        acc += __shfl_xor(acc, off, 32);
    if (lane == 0) {
        float z = acc + bc[0];
        out[n] = 1.0f / (1.0f + expf(-z));
    }
}

extern "C" void kernel_launch(void* const* d_in, const int* in_sizes, int n_in,
                              void* d_out, int out_size, void* d_ws, size_t ws_size,
                              hipStream_t stream) {
    const float* x    = (const float*)d_in[0];
    const int*   ei   = (const int*)d_in[1];
    const float* W1   = (const float*)d_in[2];
    const float* b1   = (const float*)d_in[3];
    const float* W2   = (const float*)d_in[4];
    const float* b2   = (const float*)d_in[5];
    const float* Wc   = (const float*)d_in[6];
    const float* bc   = (const float*)d_in[7];
    float*       out  = (float*)d_out;

    const int N = in_sizes[0] / 32;      // 100000
    const int E = in_sizes[1] / 2;       // 3200000
    const int* src = ei;
    const int* dst = ei + E;

    // Workspace layout: dinv [N] | bufA [N*64] | bufB [N*64]
    float* dinv = (float*)d_ws;
    float* bufA = dinv + N;
    float* bufB = bufA + (size_t)N * HIDC;

    const int TB = 256;

    // --- degrees -> dinv ---
    hipMemsetAsync(dinv, 0, (size_t)N * sizeof(float), stream);
    degree_kernel<<<(E + TB - 1) / TB, TB, 0, stream>>>(dst, dinv, E);
    dinv_kernel<<<(N + TB - 1) / TB, TB, 0, stream>>>(dinv, N);

    // --- layer 1: bufA = x @ W1 (K=32) ---
    {
        int waves  = ((N + 15) / 16) * (HIDC / 16);
        int thr    = waves * 32;
        gemm_wmma_kernel<32, HIDC><<<(thr + TB - 1) / TB, TB, 0, stream>>>(x, W1, bufA, N);
    }
    hipMemsetAsync(bufB, 0, (size_t)N * HIDC * sizeof(float), stream);
    {
        long long thr = (long long)E * 32;
        scatter_kernel<<<(int)((thr + TB - 1) / TB), TB, 0, stream>>>(src, dst, dinv, bufA, bufB, E);
    }
    finalize_kernel<<<((long long)N * HIDC + TB - 1) / TB, TB, 0, stream>>>(bufA, dinv, b1, bufB, N);

    // --- layer 2: bufA = bufB @ W2 (K=64) ---
    {
        int waves = ((N + 15) / 16) * (HIDC / 16);
        int thr   = waves * 32;
        gemm_wmma_kernel<64, HIDC><<<(thr + TB - 1) / TB, TB, 0, stream>>>(bufB, W2, bufA, N);
    }
    hipMemsetAsync(bufB, 0, (size_t)N * HIDC * sizeof(float), stream);
    {
        long long thr = (long long)E * 32;
        scatter_kernel<<<(int)((thr + TB - 1) / TB), TB, 0, stream>>>(src, dst, dinv, bufA, bufB, E);
    }
    finalize_kernel<<<((long long)N * HIDC + TB - 1) / TB, TB, 0, stream>>>(bufA, dinv, b2, bufB, N);

    // --- head ---
    {
        long long thr = (long long)N * 32;
        head_kernel<<<(int)((thr + TB - 1) / TB), TB, 0, stream>>>(bufB, Wc, bc, out, N);
    }
}